// gcnNet_59098749993118
// MI455X (gfx1250) — compile-verified
//
#include <hip/hip_runtime.h>
#include <math.h>

// GCN 2-layer forward for MI455X (gfx1250, wave32).
// GEMMs use V_WMMA_F32_16X16X4_F32 (fp32 matrix pipe, full precision);
// edge propagation uses coalesced gathers + global f32 atomics (L2-resident).

#define NODES 100000
#define EDGES 3200000
#define F_IN  61
#define HID   32
#define CLS   2

typedef __attribute__((ext_vector_type(2))) float v2f;
typedef __attribute__((ext_vector_type(8))) float v8f;

__device__ __forceinline__ void atomic_add_f32(float* p, float v) {
  // relaxed, agent scope -> global_atomic_add_f32 (no return)
  __hip_atomic_fetch_add(p, v, __ATOMIC_RELAXED, __HIP_MEMORY_SCOPE_AGENT);
}

__global__ void __launch_bounds__(256) zero_f32(float* __restrict__ p, int n) {
  int i = blockIdx.x * blockDim.x + threadIdx.x;
  if (i < n) p[i] = 0.0f;
}

// one atomic per edge on dst
__global__ void __launch_bounds__(256) degree_kernel(const int* __restrict__ dst,
                                                     float* __restrict__ deg) {
  int i = blockIdx.x * blockDim.x + threadIdx.x;
  if (i < EDGES) atomic_add_f32(&deg[dst[i]], 1.0f);
}

__global__ void __launch_bounds__(256) dinv_kernel(const float* __restrict__ deg,
                                                   float* __restrict__ dinv) {
  int i = blockIdx.x * blockDim.x + threadIdx.x;
  if (i < NODES) dinv[i] = rsqrtf(deg[i] + 1.0f);  // segment_sum + 1 self-loop
}

// lin1 = X[N,61] @ W1[61,32] via V_WMMA_F32_16X16X4_F32.
// One wave -> 16x32 output tile (two 16x16 column tiles sharing A).
__global__ void __launch_bounds__(256) gemm1_wmma(const float* __restrict__ X,
                                                  const float* __restrict__ W,
                                                  float* __restrict__ H) {
  const int wave = blockIdx.x * (blockDim.x >> 5) + (threadIdx.x >> 5);
  if (wave >= NODES / 16) return;            // uniform per wave: EXEC stays all-1s
  const int lane = threadIdx.x & 31;
  const int half = lane >> 4;                // 0: K=k,k+1   1: K=k+2,k+3
  const int l16  = lane & 15;                // A: row M ; B/D: col N
  const int row  = wave * 16 + l16;

  v8f c0 = {}; v8f c1 = {};
  for (int k = 0; k < 64; k += 4) {          // K=61 padded to 64
    const int ka = k + 2 * half;
    v2f a, b0, b1;
    a.x  = (ka     < F_IN) ? X[row * F_IN + ka]     : 0.0f;
    a.y  = (ka + 1 < F_IN) ? X[row * F_IN + ka + 1] : 0.0f;
    b0.x = (ka     < F_IN) ? W[ka       * HID + l16]      : 0.0f;
    b0.y = (ka + 1 < F_IN) ? W[(ka + 1) * HID + l16]      : 0.0f;
    b1.x = (ka     < F_IN) ? W[ka       * HID + 16 + l16] : 0.0f;
    b1.y = (ka + 1 < F_IN) ? W[(ka + 1) * HID + 16 + l16] : 0.0f;
    c0 = __builtin_amdgcn_wmma_f32_16x16x4_f32(false, a, false, b0, 0, c0, false, false);
    c1 = __builtin_amdgcn_wmma_f32_16x16x4_f32(false, a, false, b1, 0, c1, false, false);
  }
#pragma unroll
  for (int j = 0; j < 8; ++j) {              // D: VGPR j = row j (+8 for hi half)
    const int m = wave * 16 + j + 8 * half;
    H[m * HID + l16]      = c0[j];
    H[m * HID + 16 + l16] = c1[j];
  }
}

// agg1[dst, c] += dinv[src]*dinv[dst] * lin1[src, c]; 32 lanes = one edge row (128B coalesced)
__global__ void __launch_bounds__(256) scatter1(const int* __restrict__ ei,
                                                const float* __restrict__ dinv,
                                                const float* __restrict__ H,
                                                float* __restrict__ agg) {
  const unsigned idx = blockIdx.x * blockDim.x + threadIdx.x;
  if (idx >= (unsigned)EDGES * HID) return;
  const int e = idx >> 5, c = idx & 31;
  const int s = ei[e], d = ei[EDGES + e];
  const float nrm = dinv[s] * dinv[d];
  atomic_add_f32(&agg[d * HID + c], nrm * H[s * HID + c]);
}

// h2 = relu(agg1 + lin1*dinv^2 + b1)
__global__ void __launch_bounds__(256) post1(const float* __restrict__ agg,
                                             const float* __restrict__ lin,
                                             const float* __restrict__ dinv,
                                             const float* __restrict__ b1,
                                             float* __restrict__ out) {
  const unsigned idx = blockIdx.x * blockDim.x + threadIdx.x;
  if (idx >= (unsigned)NODES * HID) return;
  const int i = idx >> 5, c = idx & 31;
  const float di = dinv[i];
  const float v = agg[idx] + lin[idx] * di * di + b1[c];
  out[idx] = fmaxf(v, 0.0f);
}

// lin2 = H[N,32] @ W2[32,2] via WMMA f32 (cols 2..15 zero-padded, only 2 stored)
__global__ void __launch_bounds__(256) gemm2_wmma(const float* __restrict__ H,
                                                  const float* __restrict__ W,
                                                  float* __restrict__ O) {
  const int wave = blockIdx.x * (blockDim.x >> 5) + (threadIdx.x >> 5);
  if (wave >= NODES / 16) return;
  const int lane = threadIdx.x & 31;
  const int half = lane >> 4;
  const int l16  = lane & 15;
  const int row  = wave * 16 + l16;

  v8f c0 = {};
  for (int k = 0; k < HID; k += 4) {
    const int ka = k + 2 * half;
    v2f a, b;
    a.x = H[row * HID + ka];
    a.y = H[row * HID + ka + 1];
    b.x = (l16 < CLS) ? W[ka       * CLS + l16] : 0.0f;
    b.y = (l16 < CLS) ? W[(ka + 1) * CLS + l16] : 0.0f;
    c0 = __builtin_amdgcn_wmma_f32_16x16x4_f32(false, a, false, b, 0, c0, false, false);
  }
  if (l16 < CLS) {
#pragma unroll
    for (int j = 0; j < 8; ++j) {
      const int m = wave * 16 + j + 8 * half;
      O[m * CLS + l16] = c0[j];
    }
  }
}

// agg2[dst, c] += norm * lin2[src, c]  (2 channels per edge)
__global__ void __launch_bounds__(256) scatter2(const int* __restrict__ ei,
                                                const float* __restrict__ dinv,
                                                const float* __restrict__ L,
                                                float* __restrict__ agg) {
  const unsigned idx = blockIdx.x * blockDim.x + threadIdx.x;
  if (idx >= (unsigned)EDGES * CLS) return;
  const int e = idx >> 1, c = idx & 1;
  const int s = ei[e], d = ei[EDGES + e];
  const float nrm = dinv[s] * dinv[d];
  atomic_add_f32(&agg[d * CLS + c], nrm * L[s * CLS + c]);
}

// out = log_softmax(agg2 + lin2*dinv^2 + b2) over 2 classes
__global__ void __launch_bounds__(256) final_kernel(const float* __restrict__ agg,
                                                    const float* __restrict__ lin,
                                                    const float* __restrict__ dinv,
                                                    const float* __restrict__ b2,
                                                    float* __restrict__ out) {
  const int i = blockIdx.x * blockDim.x + threadIdx.x;
  if (i >= NODES) return;
  const float di = dinv[i];
  const float sl = di * di;
  const float v0 = agg[i * 2 + 0] + lin[i * 2 + 0] * sl + b2[0];
  const float v1 = agg[i * 2 + 1] + lin[i * 2 + 1] * sl + b2[1];
  const float m = fmaxf(v0, v1);
  const float lse = m + logf(expf(v0 - m) + expf(v1 - m));
  out[i * 2 + 0] = v0 - lse;
  out[i * 2 + 1] = v1 - lse;
}

static inline int cdiv(long long a, long long b) { return (int)((a + b - 1) / b); }

extern "C" void kernel_launch(void* const* d_in, const int* in_sizes, int n_in,
                              void* d_out, int out_size, void* d_ws, size_t ws_size,
                              hipStream_t stream) {
  const float* x  = (const float*)d_in[0];
  const int*   ei = (const int*)d_in[1];   // [2,E]: src=ei[0..E-1], dst=ei[E..2E-1]
  // d_in[2] = dropout (unused, eval mode)
  const float* W1 = (const float*)d_in[3];
  const float* b1 = (const float*)d_in[4];
  const float* W2 = (const float*)d_in[5];
  const float* b2 = (const float*)d_in[6];
  float* out = (float*)d_out;

  float* ws   = (float*)d_ws;
  float* deg  = ws;                         // N
  float* dinv = deg  + NODES;               // N
  float* lin1 = dinv + NODES;               // N*HID
  float* agg1 = lin1 + (size_t)NODES * HID; // N*HID
  float* h2   = agg1 + (size_t)NODES * HID; // N*HID
  float* lin2 = h2   + (size_t)NODES * HID; // N*CLS
  float* agg2 = lin2 + (size_t)NODES * CLS; // N*CLS

  const int T = 256;

  zero_f32<<<cdiv(NODES, T), T, 0, stream>>>(deg, NODES);
  zero_f32<<<cdiv((long long)NODES * HID, T), T, 0, stream>>>(agg1, NODES * HID);
  zero_f32<<<cdiv((long long)NODES * CLS, T), T, 0, stream>>>(agg2, NODES * CLS);

  degree_kernel<<<cdiv(EDGES, T), T, 0, stream>>>(ei + EDGES, deg);
  dinv_kernel<<<cdiv(NODES, T), T, 0, stream>>>(deg, dinv);

  gemm1_wmma<<<cdiv(NODES / 16, 8), T, 0, stream>>>(x, W1, lin1);
  scatter1<<<cdiv((long long)EDGES * HID, T), T, 0, stream>>>(ei, dinv, lin1, agg1);
  post1<<<cdiv((long long)NODES * HID, T), T, 0, stream>>>(agg1, lin1, dinv, b1, h2);

  gemm2_wmma<<<cdiv(NODES / 16, 8), T, 0, stream>>>(h2, W2, lin2);
  scatter2<<<cdiv((long long)EDGES * CLS, T), T, 0, stream>>>(ei, dinv, lin2, agg2);
  final_kernel<<<cdiv(NODES, T), T, 0, stream>>>(agg2, lin2, dinv, b2, out);
}